// OptimalTransportLayer_34102040330984
// MI455X (gfx1250) — compile-verified
//
#include <hip/hip_runtime.h>

// Sinkhorn (B=64, H=W=512, gamma=1, 20 iters + final row norm) in scaling-
// vector form: log_P = (rowmin_i - M_ij) + u_i + v_j.
//   row pass: u_i = LOG_R - LSE_j(rowmin_i - M_ij + v_j)
//   col pass: v_j = LOG_C - LSE_i(rowmin_i - M_ij + u_i)
// One 1024-thread workgroup per batch slice; M slice (1 MB) stays L2-resident
// across the 41 streaming passes; u/v/rowmin live in LDS. Single launch, no
// global sync. Wave-sum reduction uses V_WMMA_F32_16X16X4_F32 with a ones
// B-matrix (folds lane L with L+16 into 16 row sums).

#define HDIM 512
#define WDIM 512
#define NITER 20
#define LOG_R (-6.2383246250395077847f) /* -log(512) */
#define LOG_C (-6.2383246250395077847f) /* -log(512) */

typedef __attribute__((ext_vector_type(2))) float v2f;
typedef __attribute__((ext_vector_type(8))) float v8f;

__device__ __forceinline__ float wave_max32(float x) {
#pragma unroll
  for (int m = 1; m <= 16; m <<= 1) x = fmaxf(x, __shfl_xor(x, m, 32));
  return x;
}

__device__ __forceinline__ float wave_min32(float x) {
#pragma unroll
  for (int m = 1; m <= 16; m <<= 1) x = fminf(x, __shfl_xor(x, m, 32));
  return x;
}

// Sum of s across all 32 lanes. WMMA 16x16x4 f32 with B = ones:
// A(VGPR0)=s, A(VGPR1)=0 => D_m = s_m + s_{m+16}, replicated over all 16
// columns of D. Then 7 adds collapse the 8 D VGPRs (rows 0-7 on lanes 0-15,
// rows 8-15 on lanes 16-31) and one xor-16 shuffle merges the halves.
// Result is uniform across the wave. Requires EXEC all ones (no divergence).
__device__ __forceinline__ float wave_sum_wmma(float s) {
  v2f a;    a.x = s;    a.y = 0.0f;
  v2f ones; ones.x = 1.0f; ones.y = 1.0f;
  v8f c = {};
  v8f d = __builtin_amdgcn_wmma_f32_16x16x4_f32(
      /*neg_a=*/false, a, /*neg_b=*/false, ones,
      /*c_mod=*/(short)0, c, /*reuse_a=*/false, /*reuse_b=*/false);
  float t = ((d[0] + d[1]) + (d[2] + d[3])) + ((d[4] + d[5]) + (d[6] + d[7]));
  t += __shfl_xor(t, 16, 32);
  return t;
}

__global__ __launch_bounds__(1024)
void OptimalTransportLayer_34102040330984_kernel(const float* __restrict__ M,
                                                 float* __restrict__ out) {
  __shared__ float rowmin[HDIM];
  __shared__ float u[HDIM];
  __shared__ float v[WDIM];
  __shared__ float scm[1024];
  __shared__ float scs[1024];

  const int tid  = threadIdx.x;
  const int lane = tid & 31;
  const int wave = tid >> 5;
  const float* __restrict__ Mb = M + (size_t)blockIdx.x * (HDIM * WDIM);
  float* __restrict__ Ob = out + (size_t)blockIdx.x * (HDIM * WDIM);

  // ---- Phase 0: row minima (wave-per-row, 16 rows per wave) ----
#pragma unroll 1
  for (int chunk = 0; chunk < 16; ++chunk) {
    const int row = chunk * 32 + wave;
    const float* __restrict__ mr = Mb + row * WDIM;
    float mn = __builtin_inff();
#pragma unroll
    for (int k = 0; k < 16; ++k) mn = fminf(mn, mr[k * 32 + lane]);
    mn = wave_min32(mn);
    if (lane == 0) rowmin[row] = mn;
  }
  if (tid < WDIM) v[tid] = 0.0f;
  __syncthreads();

  for (int it = 0; it <= NITER; ++it) {
    // ---- Row pass: u_i = LOG_R - LSE_j(rowmin_i - M_ij + v_j) ----
#pragma unroll 1
    for (int chunk = 0; chunk < 16; ++chunk) {
      const int row = chunk * 32 + wave;
      const float* __restrict__ mr = Mb + row * WDIM;
      const float rm = rowmin[row];
      float x[16];
      float mx = -__builtin_inff();
#pragma unroll
      for (int k = 0; k < 16; ++k) {
        const int j = k * 32 + lane;           // coalesced 128B per wave
        x[k] = rm - mr[j] + v[j];
        mx = fmaxf(mx, x[k]);
      }
      mx = wave_max32(mx);
      float s = 0.0f;
#pragma unroll
      for (int k = 0; k < 16; ++k) s += __expf(x[k] - mx);
      const float S = wave_sum_wmma(s);        // v_wmma_f32_16x16x4_f32
      if (lane == 0) u[row] = LOG_R - (mx + __logf(S));
    }
    __syncthreads();
    if (it == NITER) break;  // final extra row normalization: stop here

    // ---- Col pass: v_j = LOG_C - LSE_i(rowmin_i - M_ij + u_i) ----
    {
      const int j    = tid & (WDIM - 1);       // consecutive lanes -> coalesced
      const int half = tid >> 9;               // split H between 2 threads/col
      float m = -__builtin_inff();
      float s = 0.0f;
#pragma unroll 4
      for (int ii = 0; ii < HDIM / 2; ++ii) {
        const int i = half * (HDIM / 2) + ii;
        const float x  = rowmin[i] - Mb[i * WDIM + j] + u[i];
        const float nm = fmaxf(m, x);
        s = s * __expf(m - nm) + __expf(x - nm);  // online LSE
        m = nm;
      }
      scm[tid] = m;
      scs[tid] = s;
      __syncthreads();
      if (tid < WDIM) {
        const float m2 = scm[tid + 512], s2 = scs[tid + 512];
        const float mm = fmaxf(m, m2);
        const float ss = s * __expf(m - mm) + s2 * __expf(m2 - mm);
        v[tid] = LOG_C - (mm + __logf(ss));
      }
      __syncthreads();
    }
  }

  // ---- Output: P_ij = exp(rowmin_i - M_ij + u_i + v_j), coalesced ----
#pragma unroll 1
  for (int idx = tid; idx < HDIM * WDIM; idx += 1024) {
    const int i = idx >> 9;
    const int j = idx & (WDIM - 1);
    Ob[idx] = __expf(rowmin[i] - Mb[idx] + u[i] + v[j]);
  }
}

extern "C" void kernel_launch(void* const* d_in, const int* in_sizes, int n_in,
                              void* d_out, int out_size, void* d_ws, size_t ws_size,
                              hipStream_t stream) {
  (void)n_in; (void)out_size; (void)d_ws; (void)ws_size;
  const float* M = (const float*)d_in[0];
  float* out = (float*)d_out;
  const int B = in_sizes[0] / (HDIM * WDIM);  // 64
  OptimalTransportLayer_34102040330984_kernel<<<dim3(B), dim3(1024), 0, stream>>>(M, out);
}